// MM_GCN2_67087389164143
// MI455X (gfx1250) — compile-verified
//
#include <hip/hip_runtime.h>
#include <cmath>

typedef __attribute__((ext_vector_type(2))) float v2f;
typedef __attribute__((ext_vector_type(8))) float v8f;

#define NFEAT  1024
#define NHID   512
#define NMOD_N 3072   // N (rows per modality)
#define NROWS  9216   // 3N
#define NDIA   48
#define DIAL   64

// Padded LDS row strides (even => 8B-aligned b64 reads; chosen so that
// (stride*lane + k) mod 64 banks are all distinct across the wave).
#define PAD32  36     // for 32-wide K slabs
#define PAD64  68     // for 64-wide tiles

__device__ __forceinline__ v8f wmma4(v2f a, v2f b, v8f c) {
  // D = A(16x4,f32) * B(4x16,f32) + C(16x16,f32)
  return __builtin_amdgcn_wmma_f32_16x16x4_f32(false, a, false, b, (short)0, c,
                                               false, false);
}

// ---------------------------------------------------------------------------
// Stage 1: x = [relu(a@W0+b0); relu(v@W2+b2); relu(l@W1+b1)]   [9216 x 512]
// Block: 256 thr (8 waves) -> 128x64 output tile. K staged in 32-wide slabs.
// A staged row-major (padded); W staged TRANSPOSED (col-major, padded) so the
// B operand is a single contiguous ds_load_b64 per WMMA.
// ---------------------------------------------------------------------------
__global__ __launch_bounds__(256) void fc_relu_kernel(
    const float* __restrict__ Aa, const float* __restrict__ Av,
    const float* __restrict__ Al,
    const float* __restrict__ W0, const float* __restrict__ b0,
    const float* __restrict__ W1, const float* __restrict__ b1,
    const float* __restrict__ W2, const float* __restrict__ b2,
    float* __restrict__ X)
{
  __shared__ float sA[128 * PAD32];   // [row][k]
  __shared__ float sBt[64 * PAD32];   // [col][k]  (transposed W slab)

  const int tid  = threadIdx.x;
  const int lane = tid & 31;
  const int wave = tid >> 5;
  const int l16  = lane & 15;
  const int kh   = (lane >> 4) << 1;  // K offset 0 / 2 (A & B operand layout)
  const int rh   = (lane >> 4) << 3;  // row offset 0 / 8 (C/D layout)

  const int rowBlk  = blockIdx.y;           // 0..71 (24 per modality)
  const int m       = rowBlk / 24;
  const int rowBase = (rowBlk % 24) * 128;  // row within modality
  const int colBase = blockIdx.x * 64;

  const float* A; const float* W; const float* bias;
  if      (m == 0) { A = Aa; W = W0; bias = b0; }
  else if (m == 1) { A = Av; W = W2; bias = b2; }
  else             { A = Al; W = W1; bias = b1; }

  v8f acc[4] = {};

  for (int kb = 0; kb < NFEAT; kb += 32) {
    for (int idx = tid; idx < (128 * 32) / 4; idx += 256) {
      const int off = idx * 4;
      const int r = off >> 5, c = off & 31;
      *(float4*)&sA[r * PAD32 + c] =
          *(const float4*)&A[(size_t)(rowBase + r) * NFEAT + kb + c];
    }
    for (int idx = tid; idx < (32 * 64) / 4; idx += 256) {
      const int off = idx * 4;
      const int r = off >> 6, c = off & 63;  // r = k row, c = col
      const float4 w = *(const float4*)&W[(size_t)(kb + r) * NHID + colBase + c];
      sBt[(c + 0) * PAD32 + r] = w.x;
      sBt[(c + 1) * PAD32 + r] = w.y;
      sBt[(c + 2) * PAD32 + r] = w.z;
      sBt[(c + 3) * PAD32 + r] = w.w;
    }
    __syncthreads();
#pragma unroll
    for (int kk = 0; kk < 32; kk += 4) {
      const v2f av = *(const v2f*)&sA[(wave * 16 + l16) * PAD32 + kk + kh];
#pragma unroll
      for (int nt = 0; nt < 4; ++nt) {
        const v2f bv = *(const v2f*)&sBt[(nt * 16 + l16) * PAD32 + kk + kh];
        acc[nt] = wmma4(av, bv, acc[nt]);
      }
    }
    __syncthreads();
  }

#pragma unroll
  for (int nt = 0; nt < 4; ++nt) {
    const int col = colBase + nt * 16 + l16;
    const float bc = bias[col];
#pragma unroll
    for (int r = 0; r < 8; ++r) {
      const int row = m * NMOD_N + rowBase + wave * 16 + r + rh;
      X[(size_t)row * NHID + col] = fmaxf(acc[nt][r] + bc, 0.0f);
    }
  }
}

// ---------------------------------------------------------------------------
// Stage 2: per (modality, dialogue): sim = acos(0.99999 * Xn @ Xn^T)  [64x64]
// One block per (m,d). Row norms via LDS reduction, then WMMA over K=1024.
// B = Xn^T, so both operands read K-contiguous from the padded row-major tile.
// ---------------------------------------------------------------------------
__global__ __launch_bounds__(256) void sim_kernel(
    const float* __restrict__ Fa, const float* __restrict__ Fv,
    const float* __restrict__ Fl, float* __restrict__ Sim)
{
  __shared__ float sX[64 * PAD64];
  __shared__ float sRed[256];
  __shared__ float sN[64];

  const int tid  = threadIdx.x;
  const int lane = tid & 31, wave = tid >> 5;
  const int l16  = lane & 15;
  const int kh   = (lane >> 4) << 1;
  const int rh   = (lane >> 4) << 3;

  const int blk = blockIdx.x;  // 0..143
  const int m = blk / NDIA, d = blk % NDIA;
  const float* F  = (m == 0) ? Fa : (m == 1) ? Fv : Fl;
  const float* Fd = F + (size_t)d * DIAL * NFEAT;

  // Row 2-norms: 4 threads per row, 256 features each, LDS combine.
  {
    const int row = tid >> 2, part = tid & 3;
    const float* xr = Fd + (size_t)row * NFEAT + part * 256;
    float s = 0.f;
    for (int c = 0; c < 256; ++c) { const float t = xr[c]; s += t * t; }
    sRed[tid] = s;
  }
  __syncthreads();
  if (tid < 64)
    sN[tid] = rsqrtf(sRed[tid * 4] + sRed[tid * 4 + 1] +
                     sRed[tid * 4 + 2] + sRed[tid * 4 + 3]);
  __syncthreads();

  // 16 output tiles of 16x16; wave w owns tiles 2w, 2w+1 (same row band).
  const int rt  = wave >> 1;
  const int ct0 = (wave * 2) & 3;

  v8f acc[2] = {};

  for (int kb = 0; kb < NFEAT; kb += 64) {
    for (int idx = tid; idx < (64 * 64) / 4; idx += 256) {
      const int off = idx * 4;
      const int r = off >> 6, c = off & 63;
      float4 vv = *(const float4*)&Fd[(size_t)r * NFEAT + kb + c];
      const float nr = sN[r];
      vv.x *= nr; vv.y *= nr; vv.z *= nr; vv.w *= nr;
      *(float4*)&sX[r * PAD64 + c] = vv;
    }
    __syncthreads();
#pragma unroll
    for (int kk = 0; kk < 64; kk += 4) {
      const v2f av = *(const v2f*)&sX[(rt * 16 + l16) * PAD64 + kk + kh];
#pragma unroll
      for (int t = 0; t < 2; ++t) {
        const int ct = ct0 + t;  // B[k][j] = Xn[j][k] -> row-major read
        const v2f bv = *(const v2f*)&sX[(ct * 16 + l16) * PAD64 + kk + kh];
        acc[t] = wmma4(av, bv, acc[t]);
      }
    }
    __syncthreads();
  }

  float* Sb = Sim + (size_t)blk * (DIAL * DIAL);
#pragma unroll
  for (int t = 0; t < 2; ++t) {
    const int col = (ct0 + t) * 16 + l16;
#pragma unroll
    for (int r = 0; r < 8; ++r) {
      const int row = rt * 16 + r + rh;
      Sb[row * 64 + col] = acosf(0.99999f * acc[t][r]);
    }
  }
}

// ---------------------------------------------------------------------------
// Stage 3: dinv[r] = rsqrt(rowsum(sim block) + 2*0.99999)
// ---------------------------------------------------------------------------
__global__ __launch_bounds__(256) void dinv_kernel(const float* __restrict__ Sim,
                                                   float* __restrict__ Dinv)
{
  const int r = blockIdx.x * 256 + threadIdx.x;
  if (r >= NROWS) return;
  const int m = r / NMOD_N, rem = r % NMOD_N;
  const int d = rem >> 6, i = rem & 63;
  const float* srow = Sim + ((size_t)(m * NDIA + d) * 64 + i) * 64;
  float s = 2.0f * 0.99999f;  // two cross-modal links per utterance
  for (int j = 0; j < 64; ++j) s += srow[j];
  Dinv[r] = rsqrtf(s);
}

// ---------------------------------------------------------------------------
// Stage 4a: block-sparse propagation + support blend, with Dinv scaling of h
// fused into the staging (g = Dinv .* h never materialized):
//   sup = 0.9 * Dinv_i*(sim@(Dinv.*h) + 0.99999*((Dinv.*h)_mm1 + (Dinv.*h)_mm2))
//       + 0.1 * h0
// One block per 64-column slab x (m,d).
// ---------------------------------------------------------------------------
__global__ __launch_bounds__(256) void prop_kernel(const float* __restrict__ Sim,
                                                   const float* __restrict__ Hsrc,
                                                   const float* __restrict__ Dinv,
                                                   const float* __restrict__ H0,
                                                   float* __restrict__ Sup)
{
  __shared__ float sS[64 * PAD64];    // sim block, row-major [i][j]
  __shared__ float sGt[64 * PAD64];   // g tile TRANSPOSED: [col][k]

  const int tid  = threadIdx.x;
  const int lane = tid & 31, wave = tid >> 5;
  const int l16  = lane & 15;
  const int kh   = (lane >> 4) << 1;
  const int rh   = (lane >> 4) << 3;

  const int blk = blockIdx.y;  // 0..143
  const int m = blk / NDIA, d = blk % NDIA;
  const int colBase = blockIdx.x * 64;
  const int rowG = m * NMOD_N + d * 64;

  const float* Sb = Sim + (size_t)blk * (64 * 64);
  for (int idx = tid; idx < (64 * 64) / 4; idx += 256) {
    const int off = idx * 4;
    const int r = off >> 6, c = off & 63;
    *(float4*)&sS[r * PAD64 + c] = *(const float4*)&Sb[off];
  }
  for (int idx = tid; idx < (64 * 64) / 4; idx += 256) {
    const int off = idx * 4;
    const int r = off >> 6, c = off & 63;  // r = k (dialogue row), c = col
    const float dv = Dinv[rowG + r];
    const float4 h =
        *(const float4*)&Hsrc[(size_t)(rowG + r) * NHID + colBase + c];
    sGt[(c + 0) * PAD64 + r] = h.x * dv;
    sGt[(c + 1) * PAD64 + r] = h.y * dv;
    sGt[(c + 2) * PAD64 + r] = h.z * dv;
    sGt[(c + 3) * PAD64 + r] = h.w * dv;
  }
  __syncthreads();

  const int rt  = wave >> 1;
  const int ct0 = (wave * 2) & 3;
  v8f acc[2] = {};
#pragma unroll
  for (int kk = 0; kk < 64; kk += 4) {
    const v2f av = *(const v2f*)&sS[(rt * 16 + l16) * PAD64 + kk + kh];
#pragma unroll
    for (int t = 0; t < 2; ++t) {
      const int ct = ct0 + t;
      const v2f bv = *(const v2f*)&sGt[(ct * 16 + l16) * PAD64 + kk + kh];
      acc[t] = wmma4(av, bv, acc[t]);
    }
  }

  const int mm1 = (m + 1) % 3, mm2 = (m + 2) % 3;
#pragma unroll
  for (int t = 0; t < 2; ++t) {
    const int col = colBase + (ct0 + t) * 16 + l16;
#pragma unroll
    for (int r = 0; r < 8; ++r) {
      const int il = rt * 16 + r + rh;  // local row in dialogue block
      const int k  = d * 64 + il;       // row within modality
      const int r1 = mm1 * NMOD_N + k;
      const int r2 = mm2 * NMOD_N + k;
      const float cross =
          0.99999f * (Dinv[r1] * Hsrc[(size_t)r1 * NHID + col] +
                      Dinv[r2] * Hsrc[(size_t)r2 * NHID + col]);
      const int grow = m * NMOD_N + k;
      const float hi = Dinv[grow] * (acc[t][r] + cross);
      Sup[(size_t)grow * NHID + col] =
          0.9f * hi + 0.1f * H0[(size_t)grow * NHID + col];
    }
  }
}

// ---------------------------------------------------------------------------
// Stage 4b: h = relu(theta * (S @ Wc) + (1-theta) * S)   [9216x512 x 512x512]
// ---------------------------------------------------------------------------
__global__ __launch_bounds__(256) void layer_gemm_kernel(
    const float* __restrict__ S, const float* __restrict__ Wc,
    float* __restrict__ H, float theta, float cth)
{
  __shared__ float sA[128 * PAD32];
  __shared__ float sBt[64 * PAD32];

  const int tid  = threadIdx.x;
  const int lane = tid & 31;
  const int wave = tid >> 5;
  const int l16  = lane & 15;
  const int kh   = (lane >> 4) << 1;
  const int rh   = (lane >> 4) << 3;

  const int rowBase = blockIdx.y * 128;  // 0..9088
  const int colBase = blockIdx.x * 64;

  v8f acc[4] = {};

  for (int kb = 0; kb < NHID; kb += 32) {
    for (int idx = tid; idx < (128 * 32) / 4; idx += 256) {
      const int off = idx * 4;
      const int r = off >> 5, c = off & 31;
      *(float4*)&sA[r * PAD32 + c] =
          *(const float4*)&S[(size_t)(rowBase + r) * NHID + kb + c];
    }
    for (int idx = tid; idx < (32 * 64) / 4; idx += 256) {
      const int off = idx * 4;
      const int r = off >> 6, c = off & 63;
      const float4 w = *(const float4*)&Wc[(size_t)(kb + r) * NHID + colBase + c];
      sBt[(c + 0) * PAD32 + r] = w.x;
      sBt[(c + 1) * PAD32 + r] = w.y;
      sBt[(c + 2) * PAD32 + r] = w.z;
      sBt[(c + 3) * PAD32 + r] = w.w;
    }
    __syncthreads();
#pragma unroll
    for (int kk = 0; kk < 32; kk += 4) {
      const v2f av = *(const v2f*)&sA[(wave * 16 + l16) * PAD32 + kk + kh];
#pragma unroll
      for (int nt = 0; nt < 4; ++nt) {
        const v2f bv = *(const v2f*)&sBt[(nt * 16 + l16) * PAD32 + kk + kh];
        acc[nt] = wmma4(av, bv, acc[nt]);
      }
    }
    __syncthreads();
  }

#pragma unroll
  for (int nt = 0; nt < 4; ++nt) {
    const int col = colBase + nt * 16 + l16;
#pragma unroll
    for (int r = 0; r < 8; ++r) {
      const int row = rowBase + wave * 16 + r + rh;
      const float sv = S[(size_t)row * NHID + col];
      H[(size_t)row * NHID + col] = fmaxf(theta * acc[nt][r] + cth * sv, 0.0f);
    }
  }
}

// ---------------------------------------------------------------------------
// Stage 5: out = [l | h_a | h_v | h_l]   [3072 x 2560]
// ---------------------------------------------------------------------------
__global__ __launch_bounds__(256) void out_kernel(const float* __restrict__ L,
                                                  const float* __restrict__ H,
                                                  float* __restrict__ Out)
{
  const size_t idx = ((size_t)blockIdx.x * 256 + threadIdx.x) * 4;
  if (idx >= (size_t)NMOD_N * 2560) return;
  const int row = (int)(idx / 2560);
  const int col = (int)(idx % 2560);
  float4 vv;
  if (col < 1024)
    vv = *(const float4*)&L[(size_t)row * NFEAT + col];
  else if (col < 1536)
    vv = *(const float4*)&H[(size_t)row * NHID + (col - 1024)];
  else if (col < 2048)
    vv = *(const float4*)&H[(size_t)(NMOD_N + row) * NHID + (col - 1536)];
  else
    vv = *(const float4*)&H[(size_t)(2 * NMOD_N + row) * NHID + (col - 2048)];
  *(float4*)&Out[idx] = vv;
}

// ---------------------------------------------------------------------------
extern "C" void kernel_launch(void* const* d_in, const int* in_sizes, int n_in,
                              void* d_out, int out_size, void* d_ws,
                              size_t ws_size, hipStream_t stream)
{
  const float* a     = (const float*)d_in[0];
  const float* v     = (const float*)d_in[1];
  const float* l     = (const float*)d_in[2];
  const float* W0    = (const float*)d_in[3];
  const float* b0    = (const float*)d_in[4];
  const float* W1    = (const float*)d_in[5];
  const float* b1    = (const float*)d_in[6];
  const float* W2    = (const float*)d_in[7];
  const float* b2    = (const float*)d_in[8];
  const float* Wconv = (const float*)d_in[9];

  float* ws = (float*)d_ws;
  const size_t XSZ = (size_t)NROWS * NHID;        // 4,718,592 floats
  float* H0   = ws;
  float* H    = H0 + XSZ;
  float* SUP  = H + XSZ;
  float* SIM  = SUP + XSZ;                        // 3*48*64*64 floats
  float* DINV = SIM + (size_t)3 * NDIA * DIAL * DIAL;

  fc_relu_kernel<<<dim3(8, 72), 256, 0, stream>>>(a, v, l, W0, b0, W1, b1,
                                                  W2, b2, H0);
  sim_kernel<<<dim3(144), 256, 0, stream>>>(a, v, l, SIM);
  dinv_kernel<<<dim3(36), 256, 0, stream>>>(SIM, DINV);

  for (int i = 0; i < 8; ++i) {
    const float theta = logf(0.5f / (float)(i + 1) + 1.0f);
    prop_kernel<<<dim3(8, 144), 256, 0, stream>>>(SIM, i == 0 ? H0 : H, DINV,
                                                  H0, SUP);
    layer_gemm_kernel<<<dim3(8, 72), 256, 0, stream>>>(
        SUP, Wconv + (size_t)i * NHID * NHID, H, theta, 1.0f - theta);
  }

  out_kernel<<<dim3(7680), 256, 0, stream>>>(l, H, (float*)d_out);
}